// CorrLoss_83717502533695
// MI455X (gfx1250) — compile-verified
//
#include <hip/hip_runtime.h>
#include <hip/hip_bf16.h>

typedef float v2f __attribute__((ext_vector_type(2)));
typedef float v8f __attribute__((ext_vector_type(8)));

#define N_SAMP 4096
#define D_FEAT 512
#define LDS_STRIDE 516   // floats; 516 mod 64 == 4 -> conflict-free b64 reads
#define ROW_TILES 4      // 4 x 16 = 64 rows per block; 4x B-fragment reuse
#define COL_SPLITS 2     // column tiles split across gridDim.y
#define MARGIN_F 40.0f

// Order-preserving float -> uint encoding (monotone under unsigned compare).
__device__ __forceinline__ unsigned encf(float f) {
    unsigned u = __float_as_uint(f);
    return u ^ (unsigned)(((int)u >> 31) | (int)0x80000000);
}
__device__ __forceinline__ float decf(unsigned e) {
    unsigned u = (e & 0x80000000u) ? (e ^ 0x80000000u) : ~e;
    return __uint_as_float(u);
}

__global__ void init_minmax(unsigned* __restrict__ apEnc, unsigned* __restrict__ anEnc) {
    int i = blockIdx.x * blockDim.x + threadIdx.x;
    if (i < N_SAMP) {
        apEnc[i] = 0xFF800000u;  // encf(+inf): identity for min
        anEnc[i] = 0x007FFFFFu;  // encf(-inf): identity for max
    }
}

// Block = 64 rows (4 row tiles) x half the columns. 8 waves split column tiles.
// Each wave keeps 4 accumulator tiles so every B fragment load feeds 4 WMMAs.
__global__ __launch_bounds__(256) void gram_minmax(const float* __restrict__ feat,
                                                   const int* __restrict__ targets,
                                                   unsigned* __restrict__ apEnc,
                                                   unsigned* __restrict__ anEnc) {
    extern __shared__ float aTile[];   // 64 * LDS_STRIDE floats (~132 KB, < 320 KB/WGP)

    const int tid  = threadIdx.x;
    const int lane = tid & 31;
    const int wave = tid >> 5;       // 0..7
    const int half = lane >> 4;      // 0/1
    const int l15  = lane & 15;
    const int i0   = blockIdx.x * (16 * ROW_TILES);

    // Cooperative stage of the 64 x 512 A row-block into LDS (float4 loads).
    for (int idx = tid; idx < 16 * ROW_TILES * (D_FEAT / 4); idx += 256) {
        const int row = idx / (D_FEAT / 4);
        const int c4  = idx % (D_FEAT / 4);
        const float4 v = ((const float4*)(feat + (size_t)(i0 + row) * D_FEAT))[c4];
        float* dst = &aTile[row * LDS_STRIDE + c4 * 4];
        dst[0] = v.x; dst[1] = v.y; dst[2] = v.z; dst[3] = v.w;
    }
    __syncthreads();

    // Rows covered by this lane's accumulators: i0 + 16*t + r + 8*half.
    int ti[ROW_TILES][8];
    #pragma unroll
    for (int t = 0; t < ROW_TILES; ++t)
        #pragma unroll
        for (int r = 0; r < 8; ++r)
            ti[t][r] = targets[i0 + 16 * t + r + 8 * half];

    // A fragment base: lane L holds row (L&15), K offset 2*(L>=16).
    const float* aBase = &aTile[l15 * LDS_STRIDE + 2 * half];

    float runMin[ROW_TILES][8], runMax[ROW_TILES][8];
    #pragma unroll
    for (int t = 0; t < ROW_TILES; ++t)
        #pragma unroll
        for (int r = 0; r < 8; ++r) {
            runMin[t][r] = __builtin_inff();
            runMax[t][r] = -__builtin_inff();
        }

    const int jtPerSplit = (N_SAMP / 16) / COL_SPLITS;     // 128 column tiles per half
    const int jtBase     = blockIdx.y * jtPerSplit;

    for (int j = wave; j < jtPerSplit; j += 8) {
        const int j0 = (jtBase + j) * 16;
        const float* bBase = feat + (size_t)(j0 + l15) * D_FEAT + 2 * half;

        v8f c[ROW_TILES];
        #pragma unroll
        for (int t = 0; t < ROW_TILES; ++t)
            c[t] = (v8f){0.f, 0.f, 0.f, 0.f, 0.f, 0.f, 0.f, 0.f};

        #pragma unroll 4
        for (int k0 = 0; k0 < D_FEAT; k0 += 4) {
            const v2f b = *(const v2f*)(bBase + k0);       // global_load_b64, reused 4x
            #pragma unroll
            for (int t = 0; t < ROW_TILES; ++t) {
                const v2f a = *(const v2f*)(aBase + t * 16 * LDS_STRIDE + k0);  // ds_load_b64
                c[t] = __builtin_amdgcn_wmma_f32_16x16x4_f32(
                           false, a, false, b, (short)0, c[t], false, false);
            }
        }

        const int tj = targets[j0 + l15];
        #pragma unroll
        for (int t = 0; t < ROW_TILES; ++t)
            #pragma unroll
            for (int r = 0; r < 8; ++r) {
                const float v = c[t][r];
                if (ti[t][r] == tj) runMin[t][r] = fminf(runMin[t][r], v);
                else                runMax[t][r] = fmaxf(runMax[t][r], v);
            }
    }

    #pragma unroll
    for (int t = 0; t < ROW_TILES; ++t)
        #pragma unroll
        for (int r = 0; r < 8; ++r) {
            const int row = i0 + 16 * t + r + 8 * half;
            atomicMin(&apEnc[row], encf(runMin[t][r]));   // global_atomic_min_u32
            atomicMax(&anEnc[row], encf(runMax[t][r]));   // global_atomic_max_u32
        }
}

__global__ __launch_bounds__(1024) void finalize_loss(const unsigned* __restrict__ apEnc,
                                                      const unsigned* __restrict__ anEnc,
                                                      float* __restrict__ out) {
    __shared__ float sdata[1024];
    float s = 0.f;
    for (int i = threadIdx.x; i < N_SAMP; i += 1024) {
        const float ap = decf(apEnc[i]);
        const float an = decf(anEnc[i]);
        const float t  = an - ap + MARGIN_F;
        s += (t > 0.f) ? t : 0.f;
    }
    sdata[threadIdx.x] = s;
    __syncthreads();
    for (int off = 512; off > 0; off >>= 1) {
        if ((int)threadIdx.x < off) sdata[threadIdx.x] += sdata[threadIdx.x + off];
        __syncthreads();
    }
    if (threadIdx.x == 0) out[0] = sdata[0] * (1.0f / (float)N_SAMP);
}

extern "C" void kernel_launch(void* const* d_in, const int* in_sizes, int n_in,
                              void* d_out, int out_size, void* d_ws, size_t ws_size,
                              hipStream_t stream) {
    const float* feat    = (const float*)d_in[0];
    const int*   targets = (const int*)d_in[1];

    unsigned* apEnc = (unsigned*)d_ws;
    unsigned* anEnc = apEnc + N_SAMP;
    float* out = (float*)d_out;

    const size_t ldsBytes = (size_t)(16 * ROW_TILES) * LDS_STRIDE * sizeof(float);

    init_minmax<<<N_SAMP / 256, 256, 0, stream>>>(apEnc, anEnc);
    gram_minmax<<<dim3(N_SAMP / (16 * ROW_TILES), COL_SPLITS), 256, ldsBytes, stream>>>(
        feat, targets, apEnc, anEnc);
    finalize_loss<<<1, 1024, 0, stream>>>(apEnc, anEnc, out);
}